// EMDloss_11897059410078
// MI455X (gfx1250) — compile-verified
//
#include <hip/hip_runtime.h>
#include <math.h>

// ---------------------------------------------------------------------------
// Sinkhorn EMD on MI455X (gfx1250, wave32).
// C_ij = |x_i|^2 + |y_j|^2 - 2 x_i.y_j ; the -2x.y term is computed with
// V_WMMA_F32_16X16X4_F32 (fp32 WMMA, K=4 with K3 zero-padded). Point clouds
// (96 KB) stay cache-resident; the 134 MB cost matrix is never materialized.
// Transcendental (v_exp_f32) work dominates; WMMA co-executes with it.
// ---------------------------------------------------------------------------

typedef __attribute__((ext_vector_type(2))) float v2f;
typedef __attribute__((ext_vector_type(8))) float v8f;

#define BATCH 2
#define NPTS 4096
#define BN (BATCH * NPTS)          // 8192 points per cloud set
#define ITERS 20
#define EPS_F 0.1f
#define INV_EPS 10.0f
#define STRIPS (BN / 16)           // 512 16-row wave strips
#define WAVES_PER_BLOCK 4
#define SWEEP_BLOCKS (STRIPS / WAVES_PER_BLOCK)   // 128 blocks x 128 threads
#define LOG_MARG (-8.317766166719343f)            // -log(4096)

// --- init: squared norms, zero duals ---------------------------------------
__global__ __launch_bounds__(256) void emd_init(
    const float* __restrict__ X, const float* __restrict__ Y,
    float* __restrict__ xnorm, float* __restrict__ ynorm,
    float* __restrict__ f, float* __restrict__ g)
{
    int i = blockIdx.x * blockDim.x + threadIdx.x;
    if (i < BN) {
        const float* xp = X + (size_t)i * 3;
        xnorm[i] = xp[0] * xp[0] + xp[1] * xp[1] + xp[2] * xp[2];
        const float* yp = Y + (size_t)i * 3;
        ynorm[i] = yp[0] * yp[0] + yp[1] * yp[1] + yp[2] * yp[2];
        f[i] = 0.0f;
        g[i] = 0.0f;
    }
}

// --- one logsumexp sweep: out_i = EPS*(logmarg - LSE_j((dual_j - C_ij)/EPS))
// Called with (X,Y) for the f-update and (Y,X) for the g-update.
__global__ __launch_bounds__(128) void sinkhorn_lse(
    const float* __restrict__ Prow, const float* __restrict__ Pcol,
    const float* __restrict__ nrow, const float* __restrict__ ncol,
    const float* __restrict__ dual, float* __restrict__ outrow)
{
    const int wave = blockIdx.x * WAVES_PER_BLOCK + (threadIdx.x >> 5);
    const int lane = threadIdx.x & 31;
    const int half = lane >> 4;          // 0: lanes 0-15, 1: lanes 16-31
    const int t    = lane & 15;
    const int b    = wave / (NPTS / 16);
    const int row0 = (wave % (NPTS / 16)) * 16;

    // A fragment (16x4 fp32, rows row0..row0+15, pre-scaled by -2).
    // Layout: lanes 0-15 -> {K0,K1}, lanes 16-31 -> {K2,K3(=0 pad)}.
    const float* xp = Prow + ((size_t)(b * NPTS + row0 + t)) * 3;
    v2f a;
    a.x = -2.0f * (half ? xp[2] : xp[0]);
    a.y = half ? 0.0f : (-2.0f * xp[1]);

    // Row norms for the 8 D rows this lane holds: r = v + half*8.
    float xn[8];
#pragma unroll
    for (int v = 0; v < 8; ++v)
        xn[v] = nrow[b * NPTS + row0 + v + half * 8];

    float mAcc[8], sAcc[8];
#pragma unroll
    for (int v = 0; v < 8; ++v) { mAcc[v] = -INFINITY; sAcc[v] = 0.0f; }

#pragma unroll 2
    for (int n0 = 0; n0 < NPTS; n0 += 16) {
        const int n = n0 + t;
        const float* yp = Pcol + ((size_t)(b * NPTS + n)) * 3;
        v2f bb;
        bb.x = half ? yp[2] : yp[0];
        bb.y = half ? 0.0f : yp[1];
        const float gval = dual[b * NPTS + n];
        const float yn   = ncol[b * NPTS + n];

        v8f c = {};
        // D = (-2 X) . Y^T  for this 16x16 tile
        v8f d = __builtin_amdgcn_wmma_f32_16x16x4_f32(
            false, a, false, bb, (short)0, c, false, false);

        const float addc = (gval - yn) * INV_EPS;  // (g - |y|^2)/eps
#pragma unroll
        for (int v = 0; v < 8; ++v) {
            // val = (g - C)/eps = (g - yn - xn - d)/eps
            float val  = addc - (d[v] + xn[v]) * INV_EPS;
            float mNew = fmaxf(mAcc[v], val);
            sAcc[v] = sAcc[v] * __expf(mAcc[v] - mNew) + __expf(val - mNew);
            mAcc[v] = mNew;
        }
    }

    // Merge partial logsumexps across the 16 lanes of each half-wave.
#pragma unroll
    for (int mask = 1; mask < 16; mask <<= 1) {
#pragma unroll
        for (int v = 0; v < 8; ++v) {
            float mo = __shfl_xor(mAcc[v], mask, 32);
            float so = __shfl_xor(sAcc[v], mask, 32);
            float mNew = fmaxf(mAcc[v], mo);
            sAcc[v] = sAcc[v] * __expf(mAcc[v] - mNew) + so * __expf(mo - mNew);
            mAcc[v] = mNew;
        }
    }

    if (t == 0) {
#pragma unroll
        for (int v = 0; v < 8; ++v) {
            float lse = mAcc[v] + __logf(sAcc[v]);
            outrow[b * NPTS + row0 + v + half * 8] = EPS_F * (LOG_MARG - lse);
        }
    }
}

// --- final: per-wave partial of sum_ij exp((f_i+g_j-C_ij)/eps) * C_ij ------
__global__ __launch_bounds__(128) void emd_plan_cost(
    const float* __restrict__ Prow, const float* __restrict__ Pcol,
    const float* __restrict__ nrow, const float* __restrict__ ncol,
    const float* __restrict__ f, const float* __restrict__ g,
    float* __restrict__ partial)
{
    const int wave = blockIdx.x * WAVES_PER_BLOCK + (threadIdx.x >> 5);
    const int lane = threadIdx.x & 31;
    const int half = lane >> 4;
    const int t    = lane & 15;
    const int b    = wave / (NPTS / 16);
    const int row0 = (wave % (NPTS / 16)) * 16;

    const float* xp = Prow + ((size_t)(b * NPTS + row0 + t)) * 3;
    v2f a;
    a.x = -2.0f * (half ? xp[2] : xp[0]);
    a.y = half ? 0.0f : (-2.0f * xp[1]);

    float xn[8], frow[8];
#pragma unroll
    for (int v = 0; v < 8; ++v) {
        int r   = b * NPTS + row0 + v + half * 8;
        xn[v]   = nrow[r];
        frow[v] = f[r];
    }

    float part = 0.0f;
#pragma unroll 2
    for (int n0 = 0; n0 < NPTS; n0 += 16) {
        const int n = n0 + t;
        const float* yp = Pcol + ((size_t)(b * NPTS + n)) * 3;
        v2f bb;
        bb.x = half ? yp[2] : yp[0];
        bb.y = half ? 0.0f : yp[1];
        const float gval = g[b * NPTS + n];
        const float yn   = ncol[b * NPTS + n];

        v8f c = {};
        v8f d = __builtin_amdgcn_wmma_f32_16x16x4_f32(
            false, a, false, bb, (short)0, c, false, false);
#pragma unroll
        for (int v = 0; v < 8; ++v) {
            float Cval = d[v] + xn[v] + yn;
            float tt   = (frow[v] + gval - Cval) * INV_EPS;
            part += __expf(tt) * Cval;
        }
    }

#pragma unroll
    for (int mask = 1; mask < 32; mask <<= 1)
        part += __shfl_xor(part, mask, 32);

    if (lane == 0) partial[wave] = part;
}

// --- deterministic fixed-order reduction of 512 wave partials --------------
__global__ __launch_bounds__(256) void emd_finalize(
    const float* __restrict__ partial, float* __restrict__ out)
{
    __shared__ float sm[256];
    int t = threadIdx.x;
    sm[t] = partial[t] + partial[t + 256];
    __syncthreads();
    for (int s = 128; s > 0; s >>= 1) {
        if (t < s) sm[t] += sm[t + s];
        __syncthreads();
    }
    if (t == 0) out[0] = sm[0] * (1.0f / (float)BATCH);
}

extern "C" void kernel_launch(void* const* d_in, const int* in_sizes, int n_in,
                              void* d_out, int out_size, void* d_ws, size_t ws_size,
                              hipStream_t stream)
{
    (void)in_sizes; (void)n_in; (void)out_size; (void)ws_size;
    const float* X = (const float*)d_in[0];   // pc_gen [2,4096,3]
    const float* Y = (const float*)d_in[1];   // pc_gt  [2,4096,3]

    float* ws      = (float*)d_ws;            // ~133 KB of scratch
    float* xnorm   = ws;
    float* ynorm   = ws + BN;
    float* f       = ws + 2 * BN;
    float* g       = ws + 3 * BN;
    float* partial = ws + 4 * BN;             // 512 floats

    emd_init<<<(BN + 255) / 256, 256, 0, stream>>>(X, Y, xnorm, ynorm, f, g);

    for (int it = 0; it < ITERS; ++it) {
        // f-update: rows = X, reduce over Y columns using current g
        sinkhorn_lse<<<SWEEP_BLOCKS, WAVES_PER_BLOCK * 32, 0, stream>>>(
            X, Y, xnorm, ynorm, g, f);
        // g-update: symmetric, swap roles
        sinkhorn_lse<<<SWEEP_BLOCKS, WAVES_PER_BLOCK * 32, 0, stream>>>(
            Y, X, ynorm, xnorm, f, g);
    }

    emd_plan_cost<<<SWEEP_BLOCKS, WAVES_PER_BLOCK * 32, 0, stream>>>(
        X, Y, xnorm, ynorm, f, g, partial);
    emd_finalize<<<1, 256, 0, stream>>>(partial, (float*)d_out);
}